// _split_decision_38740605010081
// MI455X (gfx1250) — compile-verified
//
#include <hip/hip_runtime.h>

typedef float v2f __attribute__((ext_vector_type(2)));
typedef float v8f __attribute__((ext_vector_type(8)));

#define MAX_BIN 256
#define NFEAT   64
#define FG      16      // features per histogram block
#define CHUNK   16384   // samples per histogram block

// ---------------------------------------------------------------------------
// Kernel 1: zero the histogram workspace (d_ws is poisoned 0xAA)
// ---------------------------------------------------------------------------
__global__ void _pgbm_zero_kernel(float* __restrict__ ws, int n) {
    int i = blockIdx.x * blockDim.x + threadIdx.x;
    if (i < n) ws[i] = 0.0f;
}

// ---------------------------------------------------------------------------
// Kernel 2: privatized LDS histograms (ds_add_f32), flush via
// global_atomic_add_f32. Memory-bound: streams X (512 MB) once, coalesced
// int4 loads + global_prefetch_b8 ahead of the stream.
// ---------------------------------------------------------------------------
__global__ void _pgbm_hist_kernel(const int*   __restrict__ X,
                                  const float* __restrict__ grad,
                                  const float* __restrict__ hess,
                                  float* __restrict__ hist_g,
                                  float* __restrict__ hist_h,
                                  int n) {
    __shared__ float lds_g[FG][MAX_BIN];
    __shared__ float lds_h[FG][MAX_BIN];

    const int tid   = threadIdx.x;
    const int feat0 = blockIdx.y * FG;

    for (int idx = tid; idx < FG * MAX_BIN; idx += blockDim.x) {
        (&lds_g[0][0])[idx] = 0.0f;
        (&lds_h[0][0])[idx] = 0.0f;
    }
    __syncthreads();

    const int c    = tid & 3;    // which int4 (4 features) within the 16
    const int srow = tid >> 2;   // 0..63 sample lane
    const int base = blockIdx.x * CHUNK;
    const int end  = min(base + CHUNK, n);

    for (int i = base + srow; i < end; i += 64) {
        const int4 bins = *(const int4*)(X + (size_t)i * NFEAT + feat0 + c * 4);
        // prefetch next iteration's cacheline (speculative, OOB-safe)
        __builtin_prefetch(X + (size_t)(i + 64) * NFEAT + feat0 + c * 4, 0, 1);
        const float g = grad[i];
        const float h = hess[i];
        const int fl = c * 4;
        unsafeAtomicAdd(&lds_g[fl + 0][bins.x], g);
        unsafeAtomicAdd(&lds_g[fl + 1][bins.y], g);
        unsafeAtomicAdd(&lds_g[fl + 2][bins.z], g);
        unsafeAtomicAdd(&lds_g[fl + 3][bins.w], g);
        unsafeAtomicAdd(&lds_h[fl + 0][bins.x], h);
        unsafeAtomicAdd(&lds_h[fl + 1][bins.y], h);
        unsafeAtomicAdd(&lds_h[fl + 2][bins.z], h);
        unsafeAtomicAdd(&lds_h[fl + 3][bins.w], h);
    }
    __syncthreads();

    for (int idx = tid; idx < FG * MAX_BIN; idx += blockDim.x) {
        const int f = idx >> 8;
        const int b = idx & (MAX_BIN - 1);
        unsafeAtomicAdd(&hist_g[(feat0 + f) * MAX_BIN + b], lds_g[f][b]);
        unsafeAtomicAdd(&hist_h[(feat0 + f) * MAX_BIN + b], lds_h[f][b]);
    }
}

// ---------------------------------------------------------------------------
// Kernel 3: cumsum over bins as an exact f32 WMMA matmul:
//   Gl[f,b] = sum_k hist[f,k] * T[k,b],  T[k,b] = (k <= b) ? 1 : 0
// v_wmma_f32_16x16x4_f32 with f32 A/B: multiply-by-1.0 is bit-exact.
// One wave (32 threads) per 16x16 output tile; K-loop trimmed to k <= b_max.
// ---------------------------------------------------------------------------
__global__ void _pgbm_cumsum_wmma_kernel(const float* __restrict__ ws,
                                         float* __restrict__ out) {
    // 128 blocks: bit0 = matrix (g/h), bits1..6 = tile (4 ftiles x 16 ntiles)
    const int mat   = blockIdx.x & 1;
    const int tile  = blockIdx.x >> 1;
    const int ftile = tile >> 4;         // 0..3
    const int ntile = tile & 15;         // 0..15
    const int f0    = ftile * 16;
    const int nbase = ntile * 16;

    const float* __restrict__ hist = ws  + mat * (NFEAT * MAX_BIN);
    float*       __restrict__ dst  = out + mat * (NFEAT * MAX_BIN);

    const int lane  = threadIdx.x;       // 0..31
    const int m     = lane & 15;         // A-matrix row for this lane
    const int khalf = lane >> 4;         // 0: K pair {0,1}; 1: K pair {2,3}
    const int nl    = lane & 15;         // B/D column for this lane

    v8f c = {};                          // 16x16 f32 accumulator tile

    const int kend = nbase + 16;         // only k <= nbase+15 contributes
    const float* arow = hist + (size_t)(f0 + m) * MAX_BIN;

    for (int kbase = 0; kbase < kend; kbase += 4) {
        // A fragment: A[m][kbase + khalf*2 + j], even index -> 8B aligned
        v2f a = *(const v2f*)(arow + kbase + khalf * 2);
        // B fragment: triangular ones T[k][n]
        v2f b;
        b.x = ((kbase + khalf * 2 + 0) <= (nbase + nl)) ? 1.0f : 0.0f;
        b.y = ((kbase + khalf * 2 + 1) <= (nbase + nl)) ? 1.0f : 0.0f;
        // D = A*B + C   (8 args: neg_a, A, neg_b, B, c_mod, C, reuse_a, reuse_b)
        c = __builtin_amdgcn_wmma_f32_16x16x4_f32(
                false, a, false, b, (short)0, c, false, false);
    }

    // C/D layout: VGPR r -> M = r (lanes 0-15) or r+8 (lanes 16-31), N = nl
#pragma unroll
    for (int r = 0; r < 8; ++r) {
        const int mm = r + 8 * khalf;
        dst[(size_t)(f0 + mm) * MAX_BIN + nbase + nl] = c[r];
    }
}

// ---------------------------------------------------------------------------
extern "C" void kernel_launch(void* const* d_in, const int* in_sizes, int n_in,
                              void* d_out, int out_size, void* d_ws, size_t ws_size,
                              hipStream_t stream) {
    const int*   X    = (const int*)d_in[0];     // [N, 64] int32 bins
    const float* grad = (const float*)d_in[1];   // [N] f32
    const float* hess = (const float*)d_in[2];   // [N] f32
    const int n = in_sizes[1];                   // N_SAMPLES

    float* hist   = (float*)d_ws;                        // [2][64][256] f32
    float* hist_g = hist;
    float* hist_h = hist + NFEAT * MAX_BIN;
    const int hist_elems = 2 * NFEAT * MAX_BIN;          // 32768

    _pgbm_zero_kernel<<<(hist_elems + 255) / 256, 256, 0, stream>>>(hist, hist_elems);

    dim3 hgrid((n + CHUNK - 1) / CHUNK, NFEAT / FG);     // (123, 4)
    _pgbm_hist_kernel<<<hgrid, 256, 0, stream>>>(X, grad, hess, hist_g, hist_h, n);

    // 2 matrices x 4 feature-tiles x 16 bin-tiles = 128 single-wave blocks
    _pgbm_cumsum_wmma_kernel<<<128, 32, 0, stream>>>(hist, (float*)d_out);
}